// KmaxLayer_32246614458534
// MI455X (gfx1250) — compile-verified
//
#include <hip/hip_runtime.h>

// ---------------------------------------------------------------------------
// Top-3 keep + renormalize along rows of length 512.
//   x: [128, 512, 512] f32  ->  out same shape.
// Memory-bound (256 MB total @ 23.3 TB/s ~ 11 us floor). One wave32 per row.
//
// CDNA5 data pipeline per wave (grid-stride over rows):
//   stage 2 ahead : HBM -> L2   via global_prefetch_b8
//   stage 1 ahead : L2  -> LDS  via global_load_async_to_lds_b128 (ASYNCcnt),
//                   double-buffered, zero destination VGPRs
//   current       : LDS -> VGPR via ds_load_b128, top-3 via wave32 butterfly,
//                   NT global_store_b128 (write-once stream, keep out of L2)
// ---------------------------------------------------------------------------

typedef __attribute__((ext_vector_type(4))) float f4;

#define ROW_LEN 512            // floats per row
#define ROW_BYTES 2048
#define LANES 32
#define WAVES_PER_BLOCK 8

// Insert v into sorted triple t1 >= t2 >= t3 (top-3 of a multiset, counting
// duplicates -- matches top_k semantics for the k-th value).
__device__ __forceinline__ void ins3(float v, float& t1, float& t2, float& t3) {
  if (v > t1)      { t3 = t2; t2 = t1; t1 = v; }
  else if (v > t2) { t3 = t2; t2 = v; }
  else if (v > t3) { t3 = v; }
}

// Async-copy one 2KB row (global, 16B-aligned) into LDS at byte offset
// lds_base. Each lane moves 16B per instruction; INST_OFFSET is added to BOTH
// the LDS and global addresses (CDNA5 ISA), so one address
// setup covers the row with offsets 0/512/1024/1536.
__device__ __forceinline__ void async_row_to_lds(const float* __restrict__ grow,
                                                 float* lds_row, int lane) {
#if defined(__HIP_DEVICE_COMPILE__)
  unsigned lds_addr =
      (unsigned)(size_t)(__attribute__((address_space(3))) float*)lds_row +
      (unsigned)(lane * 16);
  const char* g = (const char*)grow + lane * 16;
  asm volatile(
      "global_load_async_to_lds_b128 %0, %1, off\n\t"
      "global_load_async_to_lds_b128 %0, %1, off offset:512\n\t"
      "global_load_async_to_lds_b128 %0, %1, off offset:1024\n\t"
      "global_load_async_to_lds_b128 %0, %1, off offset:1536"
      :
      : "v"(lds_addr), "v"(g)
      : "memory");
#else
  // Host pass / fallback: plain copy (never executed on device path).
  for (int k = 0; k < 4; ++k)
    ((f4*)lds_row)[k * LANES + lane] = ((const f4*)grow)[k * LANES + lane];
#endif
}

__device__ __forceinline__ void wait_async_le4() {
#if defined(__HIP_DEVICE_COMPILE__)
  asm volatile("s_wait_asynccnt 0x4" ::: "memory");
#endif
}
__device__ __forceinline__ void wait_async_le0() {
#if defined(__HIP_DEVICE_COMPILE__)
  asm volatile("s_wait_asynccnt 0x0" ::: "memory");
#endif
}

__global__ __launch_bounds__(256) void kmax3_norm_kernel(
    const float* __restrict__ x, float* __restrict__ out, int nrows) {
  // Per-wave double buffer: 8 waves * 2 bufs * 2KB = 32KB / block.
  __shared__ __align__(16) float smem[WAVES_PER_BLOCK * 2 * ROW_LEN];

  const int lane       = threadIdx.x & (LANES - 1);
  const int wave       = threadIdx.x >> 5;
  const int waveGlobal = blockIdx.x * WAVES_PER_BLOCK + wave;
  const int waveStride = gridDim.x * WAVES_PER_BLOCK;

  float* buf[2] = {&smem[(wave * 2 + 0) * ROW_LEN],
                   &smem[(wave * 2 + 1) * ROW_LEN]};

  if (waveGlobal >= nrows) return;  // uniform per wave; no barriers used

  // Prime the pipeline: stage row 0 into LDS buffer 0.
  async_row_to_lds(x + (size_t)waveGlobal * ROW_LEN, buf[0], lane);

  int cur = 0;
  for (int row = waveGlobal; row < nrows; row += waveStride) {
    const int nxt = row + waveStride;
    if (nxt < nrows) {
      // Issue next row into the other buffer, then wait until only those 4
      // async ops remain outstanding -> current row's 4 have completed
      // (async loads complete in order).
      async_row_to_lds(x + (size_t)nxt * ROW_LEN, buf[cur ^ 1], lane);
      wait_async_le4();
    } else {
      wait_async_le0();
    }

    // HBM -> L2 prefetch, two rows ahead (32 lanes x 64B covers 2KB).
    const int pf = row + 2 * waveStride;
    if (pf < nrows)
      __builtin_prefetch(x + (size_t)pf * ROW_LEN + lane * 16, 0, 1);

    // --- LDS -> VGPR: 4 x ds_load_b128 per lane ---
    const f4* __restrict__ lp = (const f4*)buf[cur];
    f4 a = lp[lane];
    f4 b = lp[LANES + lane];
    f4 c = lp[2 * LANES + lane];
    f4 d = lp[3 * LANES + lane];

    // --- per-lane top-3 over 16 values ---
    float t1 = -__builtin_inff();
    float t2 = t1, t3 = t1;
    ins3(a.x, t1, t2, t3); ins3(a.y, t1, t2, t3); ins3(a.z, t1, t2, t3); ins3(a.w, t1, t2, t3);
    ins3(b.x, t1, t2, t3); ins3(b.y, t1, t2, t3); ins3(b.z, t1, t2, t3); ins3(b.w, t1, t2, t3);
    ins3(c.x, t1, t2, t3); ins3(c.y, t1, t2, t3); ins3(c.z, t1, t2, t3); ins3(c.w, t1, t2, t3);
    ins3(d.x, t1, t2, t3); ins3(d.y, t1, t2, t3); ins3(d.z, t1, t2, t3); ins3(d.w, t1, t2, t3);

    // --- wave32 butterfly merge of sorted triples -> global top-3 ---
#pragma unroll
    for (int off = 16; off >= 1; off >>= 1) {
      float o1 = __shfl_xor(t1, off, LANES);
      float o2 = __shfl_xor(t2, off, LANES);
      float o3 = __shfl_xor(t3, off, LANES);
      ins3(o1, t1, t2, t3);
      ins3(o2, t1, t2, t3);
      ins3(o3, t1, t2, t3);
    }
    const float kth = t3;  // 3rd-largest value in the row (with duplicates)

    // --- masked sum of kept entries (x >= kth), wave32 reduce ---
    float s = 0.0f;
#define ACC3(e) s += ((e) >= kth) ? (e) : 0.0f
    ACC3(a.x); ACC3(a.y); ACC3(a.z); ACC3(a.w);
    ACC3(b.x); ACC3(b.y); ACC3(b.z); ACC3(b.w);
    ACC3(c.x); ACC3(c.y); ACC3(c.z); ACC3(c.w);
    ACC3(d.x); ACC3(d.y); ACC3(d.z); ACC3(d.w);
#undef ACC3
#pragma unroll
    for (int off = 16; off >= 1; off >>= 1) s += __shfl_xor(s, off, LANES);
    const float inv = 1.0f / s;

    // --- scale kept entries, zero the rest; NT b128 stores ---
#define SCL3(e) e = ((e) >= kth) ? (e) * inv : 0.0f
    SCL3(a.x); SCL3(a.y); SCL3(a.z); SCL3(a.w);
    SCL3(b.x); SCL3(b.y); SCL3(b.z); SCL3(b.w);
    SCL3(c.x); SCL3(c.y); SCL3(c.z); SCL3(c.w);
    SCL3(d.x); SCL3(d.y); SCL3(d.z); SCL3(d.w);
#undef SCL3
    f4* __restrict__ op = (f4*)(out + (size_t)row * ROW_LEN);
    __builtin_nontemporal_store(a, &op[lane]);
    __builtin_nontemporal_store(b, &op[LANES + lane]);
    __builtin_nontemporal_store(c, &op[2 * LANES + lane]);
    __builtin_nontemporal_store(d, &op[3 * LANES + lane]);

    cur ^= 1;
  }
}

extern "C" void kernel_launch(void* const* d_in, const int* in_sizes, int n_in,
                              void* d_out, int out_size, void* d_ws, size_t ws_size,
                              hipStream_t stream) {
  const float* x = (const float*)d_in[0];
  float* out     = (float*)d_out;

  const int nrows = in_sizes[0] / ROW_LEN;  // 128 * 512 = 65536

  int blocks = 1024;  // 8192 waves, 8 rows per wave (grid-stride)
  const int maxBlocks = (nrows + WAVES_PER_BLOCK - 1) / WAVES_PER_BLOCK;
  if (blocks > maxBlocks) blocks = maxBlocks;

  kmax3_norm_kernel<<<blocks, 256, 0, stream>>>(x, out, nrows);
}